// YOLOv8Loss_23630910063067
// MI455X (gfx1250) — compile-verified
//
#include <hip/hip_runtime.h>
#include <hip/hip_bf16.h>
#include <math.h>

#define NPRED 16384
#define NIMG  16
#define NGT   64
#define NCLS  80
#define IOU_THR 0.25f

typedef __attribute__((ext_vector_type(2))) float v2f;
typedef __attribute__((ext_vector_type(8))) float v8f;
typedef __attribute__((ext_vector_type(4))) unsigned int u32x4;
typedef __attribute__((ext_vector_type(8))) int i32x8;
typedef __attribute__((ext_vector_type(4))) int i32x4;

// ---------------- math helpers ----------------
__device__ __forceinline__ float4 sanitize4(float4 b) {
  float4 r;
  r.x = fminf(b.x, b.z); r.z = fmaxf(b.x, b.z);
  r.y = fminf(b.y, b.w); r.w = fmaxf(b.y, b.w);
  return r;
}

__device__ __forceinline__ float giou_pair(float4 a, float4 b) {
  float areaA = (a.z - a.x) * (a.w - a.y);
  float areaB = (b.z - b.x) * (b.w - b.y);
  float ltx = fmaxf(a.x, b.x), lty = fmaxf(a.y, b.y);
  float rbx = fminf(a.z, b.z), rby = fminf(a.w, b.w);
  float w = fmaxf(rbx - ltx, 0.f), h = fmaxf(rby - lty, 0.f);
  float inter = w * h;
  float uni = areaA + areaB - inter;
  float iou = inter / uni;
  float cw = fmaxf(a.z, b.z) - fminf(a.x, b.x);
  float ch = fmaxf(a.w, b.w) - fminf(a.y, b.y);
  float ac = cw * ch;
  return iou - (ac - uni) / ac;
}

// monotone float->ordered-uint packing with pred index in low 32 bits
__device__ __forceinline__ unsigned long long packKey(float f, unsigned idx) {
  unsigned u = __float_as_uint(f);
  u = (u & 0x80000000u) ? ~u : (u | 0x80000000u);
  return ((unsigned long long)u << 32) | (unsigned long long)idx;
}
__device__ __forceinline__ float unpackVal(unsigned long long k) {
  unsigned u = (unsigned)(k >> 32);
  u = (u & 0x80000000u) ? (u & 0x7FFFFFFFu) : ~u;
  return __uint_as_float(u);
}

__device__ __forceinline__ float bce1(float x) { return fmaxf(x, 0.f) - x + log1pf(__expf(-fabsf(x))); }
__device__ __forceinline__ float bce0(float x) { return fmaxf(x, 0.f)     + log1pf(__expf(-fabsf(x))); }

__device__ __forceinline__ float focal_term(float x, bool t) {
  float e   = __expf(-fabsf(x));
  float l   = log1pf(e);
  float bce = fmaxf(x, 0.f) + l - (t ? x : 0.f);
  float inv = 1.f / (1.f + e);
  float p   = (x >= 0.f) ? inv : e * inv;
  float pt  = t ? p : (1.f - p);
  float at  = t ? 0.25f : 0.75f;
  float om  = 1.f - pt;
  return at * om * om * bce;
}

// ---------------- CDNA5 TDM: async-load the 64x4 f32 GT tile into LDS ----------------
// Called by wave 0 only. Descriptor: 1D tensor of 256 f32 elements, tile = whole tensor.
__device__ __forceinline__ void tdm_load_gt_tile(float4* lds_dst, const float* gsrc) {
#if __has_builtin(__builtin_amdgcn_tensor_load_to_lds) && __has_builtin(__builtin_amdgcn_s_wait_tensorcnt)
  unsigned long long ga = (unsigned long long)(uintptr_t)gsrc;
  unsigned lds = (unsigned)(uintptr_t)lds_dst;
  u32x4 g0;
  g0[0] = 1u;                                               // count=1, no gather
  g0[1] = lds;                                              // lds_addr
  g0[2] = (unsigned)(ga & 0xFFFFFFFFull);                   // global_addr[31:0]
  g0[3] = (unsigned)((ga >> 32) & 0x01FFFFFFull) | (2u << 30); // global_addr[56:32] | type=2
  i32x8 g1;
  g1[0] = (int)(2u << 16);     // workgroup_mask=0, data_size=2 (4B)
  g1[1] = (int)(256u << 16);   // tensor_dim0[15:0]=256 in bits[63:48]
  g1[2] = (int)(1u << 16);     // tensor_dim0 hi=0, tensor_dim1 lo=1
  g1[3] = (int)(256u << 16);   // tensor_dim1 hi=0, tile_dim0=256
  g1[4] = 1;                   // tile_dim1=1, tile_dim2=0
  g1[5] = 256;                 // tensor_dim0_stride lo32
  g1[6] = 0;
  g1[7] = 0;
  i32x4 z4; z4[0] = 0; z4[1] = 0; z4[2] = 0; z4[3] = 0;
  i32x8 z8; z8[0] = 0; z8[1] = 0; z8[2] = 0; z8[3] = 0;
  z8[4] = 0; z8[5] = 0; z8[6] = 0; z8[7] = 0;
  // clang-23 / therock-10.0 form: (g0 u32x4, g1 i32x8, g2 i32x4, g3 i32x4, i32x8, cpol)
  __builtin_amdgcn_tensor_load_to_lds(g0, g1, z4, z4, z8, 0);
  __builtin_amdgcn_s_wait_tensorcnt((short)0);
#else
  const float4* s4 = (const float4*)gsrc;
  for (int i = (int)(threadIdx.x & 31); i < NGT; i += 32) lds_dst[i] = s4[i];
#endif
}

// ---------------- wave-level sum via V_WMMA_F32_16X16X4_F32 ----------------
// D = ones(16x4) x B(4x16): every D entry is a column sum of B, so
// sum(all 256 D entries) == 16 * sum(all 64 B entries) independent of lane layout.
__device__ __forceinline__ float wave_reduce_sum(float v) {
#if __has_builtin(__builtin_amdgcn_wmma_f32_16x16x4_f32)
  v2f a; a[0] = 1.0f; a[1] = 1.0f;
  v2f b; b[0] = v;    b[1] = 0.0f;
  v8f c = {0.f, 0.f, 0.f, 0.f, 0.f, 0.f, 0.f, 0.f};
  v8f d = __builtin_amdgcn_wmma_f32_16x16x4_f32(false, a, false, b, (short)0, c, false, false);
  float s = d[0] + d[1] + d[2] + d[3] + d[4] + d[5] + d[6] + d[7];
#pragma unroll
  for (int off = 16; off > 0; off >>= 1) s += __shfl_xor(s, off, 32);
  return s * 0.0625f;
#else
#pragma unroll
  for (int off = 16; off > 0; off >>= 1) v += __shfl_xor(v, off, 32);
  return v;
#endif
}

// ---------------- kernel 0: zero accumulators / flags / per-GT keys ----------------
__global__ void k_init(unsigned* w, int nwords) {
  for (int i = threadIdx.x; i < nwords; i += blockDim.x) w[i] = 0u;
}

// ---------------- kernel 1: per-pred best GIoU + per-GT argmax ----------------
__global__ __launch_bounds__(256) void k_match(
    const float4* __restrict__ boxp, const float4* __restrict__ gtb,
    float* __restrict__ best_iou, int* __restrict__ best_gt,
    unsigned long long* __restrict__ gt_best, unsigned* __restrict__ posAny)
{
  __shared__ float4 sgt[NGT];
  __shared__ unsigned long long skey[NGT];
  __shared__ unsigned sflag;
  const int tid = threadIdx.x;
  const int img = blockIdx.y;
  const int n   = blockIdx.x * 256 + tid;

  if (tid < 32) tdm_load_gt_tile(sgt, (const float*)(gtb + (size_t)img * NGT));
  if (tid < NGT) skey[tid] = 0ull;
  if (tid == 0) sflag = 0u;
  __syncthreads();
  if (tid < NGT) sgt[tid] = sanitize4(sgt[tid]);
  __syncthreads();

  float4 pb = sanitize4(boxp[(size_t)img * NPRED + n]);
  float bi = -1e30f; int bg = 0;
  const int m0 = tid & (NGT - 1);     // stagger start to spread LDS atomics
  for (int j = 0; j < NGT; ++j) {
    int m = (m0 + j) & (NGT - 1);
    float g = giou_pair(pb, sgt[m]);
    if (g > bi) { bi = g; bg = m; }
    unsigned long long k = packKey(g, (unsigned)n);
    if (k > skey[m]) atomicMax(&skey[m], k);   // test-then-atomic, monotone max
  }
  best_iou[(size_t)img * NPRED + n] = bi;
  best_gt [(size_t)img * NPRED + n] = bg;
  if (bi > IOU_THR) sflag = 1u;
  __syncthreads();
  if (tid < NGT) atomicMax(&gt_best[(size_t)img * NGT + tid], skey[tid]);
  if (tid == 0 && sflag) atomicOr(&posAny[img], 1u);
}

// ---------------- kernel 2: losses ----------------
__global__ __launch_bounds__(256) void k_loss(
    const float4* __restrict__ boxp, const float* __restrict__ clsp,
    const float* __restrict__ objp, const float4* __restrict__ gtb,
    const int* __restrict__ glbl,
    const float* __restrict__ best_iou, const int* __restrict__ best_gt,
    const unsigned long long* __restrict__ gt_best,
    const unsigned* __restrict__ posAny, float* __restrict__ acc)
{
  __shared__ float4 sgt[NGT];
  __shared__ int    slbl[NGT];
  __shared__ float  sgval[NGT];
  __shared__ int    sgpred[NGT];
  __shared__ unsigned s_use_fb;
  __shared__ float  wsum[8][4];

  const int tid = threadIdx.x;
  const int img = blockIdx.y;
  const int n   = blockIdx.x * 256 + tid;
  const size_t gi = (size_t)img * NPRED + n;

  if (tid < 32) tdm_load_gt_tile(sgt, (const float*)(gtb + (size_t)img * NGT));
  if (tid < NGT) {
    slbl[tid] = glbl[img * NGT + tid];
    unsigned long long k = gt_best[(size_t)img * NGT + tid];
    sgval[tid]  = unpackVal(k);
    sgpred[tid] = (int)(unsigned)(k & 0xFFFFFFFFull);
  }
  if (tid == 0) s_use_fb = (posAny[img] == 0u) ? 1u : 0u;
  __syncthreads();
  if (tid < NGT) sgt[tid] = sanitize4(sgt[tid]);
  __syncthreads();

  const float bi = best_iou[gi];
  const int   bg = best_gt[gi];
  bool pos; int mgt;
  if (s_use_fb) {                     // fallback: preds that are some GT's best
    float bv = -1e30f; int fm = -1;
    for (int m = 0; m < NGT; ++m)
      if (sgpred[m] == n && sgval[m] > bv) { bv = sgval[m]; fm = m; }
    pos = (fm >= 0); mgt = pos ? fm : 0;
  } else {
    pos = (bi > IOU_THR); mgt = bg;
  }
  const bool neg = (bi < 0.5f) && !pos;

  float box_s = 0.f, cls_s = 0.f, obj_s = 0.f, np_s = 0.f;
  const float ol = objp[gi];
  if (pos) {
    float4 pb = sanitize4(boxp[gi]);
    box_s = 1.f - giou_pair(pb, sgt[mgt]);
    np_s  = 1.f;
    obj_s = bce1(ol);
    const int lbl = slbl[mgt];
    const float4* cp4 = (const float4*)(clsp + gi * NCLS);
    __builtin_prefetch(cp4, 0, 1);
#pragma unroll
    for (int q = 0; q < NCLS / 4; ++q) {
      float4 xv = cp4[q];
      int c = q * 4;
      cls_s += focal_term(xv.x, c + 0 == lbl);
      cls_s += focal_term(xv.y, c + 1 == lbl);
      cls_s += focal_term(xv.z, c + 2 == lbl);
      cls_s += focal_term(xv.w, c + 3 == lbl);
    }
  }
  if (neg) obj_s += 0.5f * bce0(ol);

  // uniform control flow from here: EXEC all ones for WMMA
  box_s = wave_reduce_sum(box_s);
  cls_s = wave_reduce_sum(cls_s);
  obj_s = wave_reduce_sum(obj_s);
  np_s  = wave_reduce_sum(np_s);
  const int wv = tid >> 5;
  if ((tid & 31) == 0) { wsum[wv][0] = box_s; wsum[wv][1] = cls_s; wsum[wv][2] = obj_s; wsum[wv][3] = np_s; }
  __syncthreads();
  if (tid == 0) {
    float b = 0, c = 0, o = 0, p = 0;
    for (int w = 0; w < 8; ++w) { b += wsum[w][0]; c += wsum[w][1]; o += wsum[w][2]; p += wsum[w][3]; }
    atomicAdd(acc + 0, b); atomicAdd(acc + 1, c); atomicAdd(acc + 2, o); atomicAdd(acc + 3, p);
  }
}

// ---------------- kernel 3: finalize ----------------
__global__ void k_finalize(const float* __restrict__ acc, float* __restrict__ out) {
  float np = fmaxf(acc[3], 1.0f);
  out[0] = 7.5f * acc[0] / np;
  out[1] = 0.5f * acc[1] / np;
  out[2] = 1.0f * acc[2] / np;
}

// ---------------- launcher ----------------
extern "C" void kernel_launch(void* const* d_in, const int* in_sizes, int n_in,
                              void* d_out, int out_size, void* d_ws, size_t ws_size,
                              hipStream_t stream) {
  (void)in_sizes; (void)n_in; (void)out_size; (void)ws_size;
  const float4* boxp = (const float4*)d_in[0];   // [16,16384,4]
  const float*  clsp = (const float*)d_in[1];    // [16,16384,80]
  const float*  objp = (const float*)d_in[2];    // [16,16384,1]
  const float4* gtb  = (const float4*)d_in[3];   // [16,64,4]
  const int*    glbl = (const int*)d_in[4];      // [16,64]
  float* out = (float*)d_out;

  // workspace layout (~2.1 MB used)
  char* ws = (char*)d_ws;
  float*              acc     = (float*)(ws + 0);                 // 4 floats
  unsigned*           posAny  = (unsigned*)(ws + 16);             // 16 u32
  unsigned long long* gt_best = (unsigned long long*)(ws + 128);  // 16*64 u64
  float* biou = (float*)(ws + 8448);                              // 16*16384 f32
  int*   bgt  = (int*)(ws + 8448 + (size_t)NIMG * NPRED * 4);     // 16*16384 i32

  dim3 grid(NPRED / 256, NIMG), blk(256);
  k_init<<<1, 256, 0, stream>>>((unsigned*)d_ws, 8448 / 4);
  k_match<<<grid, blk, 0, stream>>>(boxp, gtb, biou, bgt, gt_best, posAny);
  k_loss <<<grid, blk, 0, stream>>>(boxp, clsp, objp, gtb, glbl, biou, bgt, gt_best, posAny, acc);
  k_finalize<<<1, 1, 0, stream>>>(acc, out);
}